// Pol2VecMulti_4870492914035
// MI455X (gfx1250) — compile-verified
//
#include <hip/hip_runtime.h>
#include <math.h>

typedef __attribute__((ext_vector_type(2))) float v2f;
typedef __attribute__((ext_vector_type(8))) float v8f;

#define BIGV   100000.0f
#define EPSV   1e-6f
#define DIMV   32
#define TILE   16
#define LROW   33          /* padded LDS row stride (floats) -> conflict-free A-layout reads */
#define BLOCK  256
#define WPB    8           /* waves per block (wave32) */
#define MAXBLK 512

__global__ __launch_bounds__(BLOCK) void ordnll_kernel(
    const float* __restrict__ z_rows,
    const float* __restrict__ z_cols,
    const float* __restrict__ gamma_rows,
    const float* __restrict__ gamma_cols,
    const float* __restrict__ bvec,
    const float* __restrict__ col_times,
    const int*   __restrict__ row_idx,
    const int*   __restrict__ col_idx,
    const int*   __restrict__ labels,
    float*       __restrict__ partials,
    int N, int ROW, int ntiles, int nwaves)
{
    __shared__ float lds[WPB][TILE * LROW];
    const int L   = threadIdx.x & 31;        // lane in wave32
    const int wib = threadIdx.x >> 5;        // wave in block
    const int gw  = blockIdx.x * WPB + wib;  // global wave id
    float* wl = lds[wib];

    const long long ostride = (long long)ROW * DIMV;   // stride between z_rows orders
    const float b0 = bvec[0], b1 = bvec[1], b2 = bvec[2], b3 = bvec[3];
    const float is2 = 0.70710678118654752440f;         // 1/sqrt(2)

    float acc = 0.0f;

    for (int tile = gw; tile < ntiles; tile += nwaves) {
        const int  pl    = L & 15;               // pair slot this lane describes
        const int  n     = tile * TILE + pl;
        const bool valid = (n < N);
        const int  nc    = valid ? n : 0;

        const int   rowp = row_idx[nc];
        const int   colp = col_idx[nc];
        const int   labp = labels[nc];
        const float tp   = col_times[colp];
        const float grp  = gamma_rows[rowp];
        const float gcp  = gamma_cols[colp];

        // ---- phase 1: coalesced gathers + Taylor combine -> diff tile in LDS ----
        for (int p = 0; p < TILE; ++p) {
            const int   prow = __shfl(rowp, p, 32);
            const int   pcol = __shfl(colp, p, 32);
            const float t    = __shfl(tp,   p, 32);
            const float c1 = t;                        // t^1 / 1!
            const float c2 = 0.5f * t * t;             // t^2 / 2!
            const float c3 = c2 * t * (1.0f / 3.0f);   // t^3 / 3!
            const float* zr = z_rows + (long long)prow * DIMV + L;   // lane = dim
            float v = zr[0];
            v = fmaf(zr[ostride],       c1, v);
            v = fmaf(zr[2 * ostride],   c2, v);
            v = fmaf(zr[3 * ostride],   c3, v);
            const float zc = z_cols[(long long)pcol * DIMV + L];
            wl[p * LROW + L] = v - zc + EPSV;          // EPS per reference, pre-square
        }
        // same-wave LDS RAW: DS ops are in-order; wait + compiler fence for safety
        asm volatile("s_wait_dscnt 0" ::: "memory");

        // ---- phase 2: dist^2 for 16 pairs = diagonal of diff * diff^T via WMMA ----
#if __has_builtin(__builtin_amdgcn_wmma_f32_16x16x4_f32)
        v8f cacc = {0.f, 0.f, 0.f, 0.f, 0.f, 0.f, 0.f, 0.f};
        // A 16x4 f32 layout: VGPR0 = K0 (lanes 0-15) / K2 (lanes 16-31); VGPR1 = K1/K3.
        // B 4x16 layout is the transpose-dual => B registers identical to A's.
        const int ko = (L < 16) ? 0 : 2;
        #pragma unroll
        for (int ch = 0; ch < 8; ++ch) {               // 8 chunks of K=4 cover DIM=32
            const int base = ch * 4;
            v2f a;
            a[0] = wl[pl * LROW + base + ko];
            a[1] = wl[pl * LROW + base + ko + 1];
            cacc = __builtin_amdgcn_wmma_f32_16x16x4_f32(
                false, a, false, a, (short)0, cacc, false, false);
        }
        // C/D layout: D[m][n] in comp m%8, lanes (m<8 ? n : n+16).
        // Diagonal D[m][m]: owner lane = m (m<8, comp m) or m+16 (m>=8, comp m-8).
        int ci = (L < 8) ? L : ((L - 24) & 7);
        float own = cacc[0];
        own = (ci == 1) ? cacc[1] : own;
        own = (ci == 2) ? cacc[2] : own;
        own = (ci == 3) ? cacc[3] : own;
        own = (ci == 4) ? cacc[4] : own;
        own = (ci == 5) ? cacc[5] : own;
        own = (ci == 6) ? cacc[6] : own;
        own = (ci == 7) ? cacc[7] : own;
        const int srcl = ((L < 8) ? L : (L + 16)) & 31;
        const float dsq = __shfl(own, srcl, 32);       // lanes 0..15: dist^2 of pair L
#else
        float dsq = 0.0f;
        for (int d = 0; d < DIMV; ++d) {
            const float v = wl[pl * LROW + d];
            dsq = fmaf(v, v, dsq);
        }
#endif

        // ---- epilogue (lanes 0..15 carry valid pairs; others masked) ----
        const float dist = sqrtf(dsq);
        const float f    = grp + gcp - dist;
        int lab = labp; lab = (lab < 1) ? 1 : ((lab > 5) ? 5 : lab);
        // theta = [-BIG, b0, b1, b2, b3, BIG]; hi = theta[lab], lo = theta[lab-1]
        const float hi = (lab >= 5) ? BIGV :
                         (lab == 1) ? b0 : (lab == 2) ? b1 : (lab == 3) ? b2 : b3;
        const float lo = (lab == 1) ? -BIGV :
                         (lab == 2) ? b0 : (lab == 3) ? b1 : (lab == 4) ? b2 : b3;
        const float phiHi = 0.5f * (1.0f + erff((hi - f) * is2));
        const float phiLo = 0.5f * (1.0f + erff((lo - f) * is2));
        const float ll = logf(phiHi - phiLo);
        acc += (valid && (L < 16)) ? (-ll) : 0.0f;
    }

    // deterministic in-wave reduction (lanes 16..31 contribute 0)
    #pragma unroll
    for (int off = 16; off >= 1; off >>= 1)
        acc += __shfl_xor(acc, off, 32);
    if (L == 0 && gw < nwaves) partials[gw] = acc;
}

__global__ __launch_bounds__(BLOCK) void reduce_kernel(
    const float* __restrict__ part, int n, float* __restrict__ out)
{
    __shared__ float s[BLOCK];
    float a = 0.0f;
    for (int i = threadIdx.x; i < n; i += BLOCK) a += part[i];   // fixed order
    s[threadIdx.x] = a;
    __syncthreads();
    for (int off = BLOCK / 2; off > 0; off >>= 1) {
        if ((int)threadIdx.x < off) s[threadIdx.x] += s[threadIdx.x + off];
        __syncthreads();
    }
    if (threadIdx.x == 0) out[0] = s[0];
}

extern "C" void kernel_launch(void* const* d_in, const int* in_sizes, int n_in,
                              void* d_out, int out_size, void* d_ws, size_t ws_size,
                              hipStream_t stream) {
    const float* z_rows     = (const float*)d_in[0];
    const float* z_cols     = (const float*)d_in[1];
    const float* gamma_rows = (const float*)d_in[2];
    const float* gamma_cols = (const float*)d_in[3];
    const float* bvec       = (const float*)d_in[4];
    const float* col_times  = (const float*)d_in[5];
    const int*   row_idx    = (const int*)d_in[6];
    const int*   col_idx    = (const int*)d_in[7];
    const int*   labels     = (const int*)d_in[8];

    const int N      = in_sizes[6];              // number of (row,col) pairs
    const int ROW    = in_sizes[2];              // gamma_rows length
    const int ntiles = (N + TILE - 1) / TILE;

    int nblocks = MAXBLK;
    const size_t maxw = ws_size / sizeof(float); // partials capacity in d_ws
    if ((size_t)nblocks * WPB > maxw) {
        nblocks = (int)(maxw / WPB);
        if (nblocks < 1) nblocks = 1;
    }
    const int nwaves = nblocks * WPB;
    float* partials = (float*)d_ws;

    ordnll_kernel<<<nblocks, BLOCK, 0, stream>>>(
        z_rows, z_cols, gamma_rows, gamma_cols, bvec, col_times,
        row_idx, col_idx, labels, partials, N, ROW, ntiles, nwaves);

    reduce_kernel<<<1, BLOCK, 0, stream>>>(partials, nwaves, (float*)d_out);
}